// PairWeightedAverage_65335042506786
// MI455X (gfx1250) — compile-verified
//
#include <hip/hip_runtime.h>
#include <hip/hip_bf16.h>

#define NN    1024
#define CL    384
#define CP    128
#define HEADS 8
#define HK    256   // HEADS*KEY_SIZE
#define COUT  384

typedef __attribute__((ext_vector_type(2))) float v2f;
typedef __attribute__((ext_vector_type(8))) float v8f;

static __device__ __forceinline__ v8f wmma_f32_4(v2f a, v2f b, v8f c) {
  // D = A(16x4 f32) * B(4x16 f32) + C(16x16 f32)
  return __builtin_amdgcn_wmma_f32_16x16x4_f32(
      /*neg_a=*/false, a, /*neg_b=*/false, b,
      /*c_mod=*/(short)0, c, /*reuse_a=*/false, /*reuse_b=*/false);
}

// -------------------- Kernel 1: LayerNorm + value projection --------------------
// value[i, o] = LN(local[i,:]) @ W_value[:, o]     (o = h*32+c)
__global__ __launch_bounds__(256) void pwa_lnvalue_kernel(
    const float* __restrict__ local, const float* __restrict__ ln_scale,
    const float* __restrict__ ln_offset, const float* __restrict__ W_value,
    float* __restrict__ value) {
  __shared__ float s_x[CL];
  __shared__ float r1[256];
  __shared__ float r2[256];
  const int i = blockIdx.x;
  const int tid = threadIdx.x;

  float s = 0.f, s2 = 0.f;
  for (int c = tid; c < CL; c += 256) {
    float x = local[(size_t)i * CL + c];
    s_x[c] = x; s += x; s2 += x * x;
  }
  r1[tid] = s; r2[tid] = s2;
  __syncthreads();
  for (int off = 128; off > 0; off >>= 1) {
    if (tid < off) { r1[tid] += r1[tid + off]; r2[tid] += r2[tid + off]; }
    __syncthreads();
  }
  const float mu   = r1[0] * (1.0f / CL);
  const float var  = r2[0] * (1.0f / CL) - mu * mu;
  const float rstd = rsqrtf(var + 1e-5f);

  for (int c = tid; c < CL; c += 256)
    s_x[c] = (s_x[c] - mu) * rstd * ln_scale[c] + ln_offset[c];
  __syncthreads();

  float acc = 0.f;
  #pragma unroll 4
  for (int c = 0; c < CL; ++c)
    acc += s_x[c] * W_value[(size_t)c * HK + tid];
  value[(size_t)i * HK + tid] = acc;
}

// -------------------- Kernel 2: fused logits + softmax + weighted average --------------------
// One workgroup per query row i. 256 threads = 8 waves.
//  Phase A: logits[j,h] = pair[i,j,:] @ W_attn  via V_WMMA_F32_16X16X4_F32, LDS-resident.
//  Phase B: masked softmax over j (per head) in LDS.
//  Phase C: out_avg[i,o] = sum_j p[j,h(o)] * value[j,o].
__global__ __launch_bounds__(256) void pwa_attn_kernel(
    const float* __restrict__ pair, const unsigned char* __restrict__ mask,
    const float* __restrict__ W_attn, const float* __restrict__ value,
    float* __restrict__ out_avg) {
  __shared__ float s_logits[NN * HEADS];   // 32 KB
  __shared__ v2f   s_B[(CP / 2) * 16];     // 8 KB: {W_attn[c], W_attn[c+1]} per col, heads padded to 16
  __shared__ float s_pm[NN];               // 4 KB: combined pair mask (mask[i] && mask[j]) as float
  __shared__ float s_red[HEADS * 32];
  __shared__ float s_stat[2 * HEADS];

  const int i    = blockIdx.x;
  const int tid  = threadIdx.x;
  const int wave = tid >> 5;
  const int lane = tid & 31;
  const int nlo  = lane & 15;   // column within 16-wide fragment
  const int half = lane >> 4;   // 0: K rows {0,1}; 1: K rows {2,3}

  // Pack B fragments: s_B[c2*16+n] = {W_attn[2*c2][n], W_attn[2*c2+1][n]}, zero for n>=8.
  for (int e = tid; e < (CP / 2) * 16; e += 256) {
    int c2 = e >> 4, n = e & 15;
    v2f b;
    if (n < HEADS) { b.x = W_attn[(2 * c2) * HEADS + n]; b.y = W_attn[(2 * c2 + 1) * HEADS + n]; }
    else           { b.x = 0.f; b.y = 0.f; }
    s_B[e] = b;
  }
  // Combined pair mask to LDS floats (branchless selects downstream, no global u8 loads in hot code).
  const bool mi = (mask[i] != 0);
  for (int j = tid; j < NN; j += 256)
    s_pm[j] = (mi && (mask[j] != 0)) ? 1.0f : 0.0f;
  __syncthreads();

  // ---- Phase A: 64 j-tiles of 16, round-robin across 8 waves ----
  for (int jt = wave; jt < NN / 16; jt += 8) {
    const float* ap = pair + ((size_t)i * NN + jt * 16 + nlo) * CP + 2 * half;
    v8f acc = {0.f, 0.f, 0.f, 0.f, 0.f, 0.f, 0.f, 0.f};
    #pragma unroll 4
    for (int k = 0; k < CP / 4; ++k) {           // K chunks of 4
      v2f a = *(const v2f*)(ap + 4 * k);         // 8B aligned (even element)
      v2f b = s_B[(2 * k + half) * 16 + nlo];
      acc = wmma_f32_4(a, b, acc);
    }
    // D: lane holds rows r+8*half, col nlo. Store cols 0..7 (real heads) with pre-softmax mask.
    if (nlo < HEADS) {
      #pragma unroll
      for (int r = 0; r < 8; ++r) {
        int j = jt * 16 + r + 8 * half;
        float pm = s_pm[j];
        s_logits[j * HEADS + nlo] = (pm != 0.0f) ? acc[r] : -1e9f;
      }
    }
  }
  __syncthreads();

  // ---- Phase B: softmax over j per head (32 threads per head) ----
  const int h   = tid & 7;
  const int grp = tid >> 3;
  float lmax = -3.4e38f;
  for (int j = grp; j < NN; j += 32) lmax = fmaxf(lmax, s_logits[j * HEADS + h]);
  s_red[h * 32 + grp] = lmax;
  __syncthreads();
  if (tid < HEADS) {
    float m = -3.4e38f;
    for (int g = 0; g < 32; ++g) m = fmaxf(m, s_red[tid * 32 + g]);
    s_stat[tid] = m;
  }
  __syncthreads();
  const float mh = s_stat[h];
  float lsum = 0.f;
  for (int j = grp; j < NN; j += 32) {
    float e = __expf(s_logits[j * HEADS + h] - mh);
    s_logits[j * HEADS + h] = e;
    lsum += e;
  }
  s_red[h * 32 + grp] = lsum;
  __syncthreads();
  if (tid < HEADS) {
    float sum = 0.f;
    for (int g = 0; g < 32; ++g) sum += s_red[tid * 32 + g];
    s_stat[HEADS + tid] = 1.0f / sum;   // sum >= 1 always (exp(max-max)=1 present)
  }
  __syncthreads();
  const float rinv = s_stat[HEADS + h];
  for (int j = grp; j < NN; j += 32)
    s_logits[j * HEADS + h] *= rinv * s_pm[j];   // normalize + post-softmax mask, branchless
  __syncthreads();

  // ---- Phase C: out_avg[i,o] = sum_j p[j, o/32] * value[j, o] ----
  const int o  = tid;
  const int oh = o >> 5;
  float acc = 0.f;
  const float* vp = value + o;
  #pragma unroll 4
  for (int j = 0; j < NN; ++j)
    acc += s_logits[j * HEADS + oh] * vp[(size_t)j * HK];
  out_avg[(size_t)i * HK + o] = acc;
}

// -------------------- Kernel 3: final projection [1024,256]@[256,384], f32 WMMA --------------------
__global__ __launch_bounds__(256) void pwa_outproj_kernel(
    const float* __restrict__ out_avg, const float* __restrict__ W_out,
    float* __restrict__ out) {
  const int tid  = threadIdx.x;
  const int wave = tid >> 5;
  const int lane = tid & 31;
  const int nlo  = lane & 15;
  const int half = lane >> 4;
  const int NT   = COUT / 16;                 // 24 col tiles
  const int tile = blockIdx.x * 8 + wave;     // 192 blocks * 8 waves = 1536 tiles
  const int mt = tile / NT;                   // i tile (0..63)
  const int nt = tile % NT;                   // col tile (0..23)

  const float* ap = out_avg + (size_t)(mt * 16 + nlo) * HK + 2 * half;
  const float* bp = W_out + (size_t)(2 * half) * COUT + nt * 16 + nlo;
  v8f acc = {0.f, 0.f, 0.f, 0.f, 0.f, 0.f, 0.f, 0.f};
  #pragma unroll 4
  for (int k = 0; k < HK / 4; ++k) {          // 64 K-chunks of 4
    v2f a = *(const v2f*)(ap + 4 * k);
    v2f b;
    b.x = bp[(size_t)(4 * k) * COUT];
    b.y = bp[(size_t)(4 * k) * COUT + COUT];
    acc = wmma_f32_4(a, b, acc);
  }
  #pragma unroll
  for (int r = 0; r < 8; ++r)
    out[(size_t)(mt * 16 + r + 8 * half) * COUT + nt * 16 + nlo] = acc[r];
}

// -------------------- launch --------------------
extern "C" void kernel_launch(void* const* d_in, const int* in_sizes, int n_in,
                              void* d_out, int out_size, void* d_ws, size_t ws_size,
                              hipStream_t stream) {
  const float*         local_in  = (const float*)d_in[0];
  const float*         pair      = (const float*)d_in[1];
  const unsigned char* mask      = (const unsigned char*)d_in[2];  // jnp bool_ -> 1 byte
  const float*         ln_scale  = (const float*)d_in[3];
  const float*         ln_offset = (const float*)d_in[4];
  const float*         W_value   = (const float*)d_in[5];
  const float*         W_attn    = (const float*)d_in[6];
  const float*         W_out     = (const float*)d_in[7];
  float* out = (float*)d_out;

  float* value   = (float*)d_ws;             // [1024, 256]  1 MB
  float* out_avg = value + (size_t)NN * HK;  // [1024, 256]  1 MB

  pwa_lnvalue_kernel<<<NN, 256, 0, stream>>>(local_in, ln_scale, ln_offset, W_value, value);
  pwa_attn_kernel<<<NN, 256, 0, stream>>>(pair, mask, W_attn, value, out_avg);
  pwa_outproj_kernel<<<(64 * (COUT / 16)) / 8, 256, 0, stream>>>(out_avg, W_out, out);
}